// GATNet_62680752718497
// MI455X (gfx1250) — compile-verified
//
#include <hip/hip_runtime.h>
#include <hip/hip_bf16.h>

// ---------------------------------------------------------------------------
// 2-layer GAT on gfx1250.
//   K_gemm0  : h0 = x @ W0            (f32 WMMA 16x16x4, 16x128 strip / wave)
//   K_nalpha0: per-node attention scalars a_dst/a_src (factored edge dots)
//   K_emax   : edge lrelu + segment-max   (uint-encoded float atomicMax)
//   K_eexp   : exp(raw - max) + segment-sum (atomicAdd)
//   K_agg0   : out0[row] += (ex/sum) * h0[col]  (float4 gather, 4 atomics/thread)
//   K_mid    : head-mean + relu -> hmid [N,64]
//   K_node1  : h1 = hmid @ W1 (64x2) + layer-1 attention scalars
//   K_emax/K_eexp (reused) : layer-1 softmax
//   K_agg1   : out1[row,h] += w * h1[col,h]
//   K_final  : d_out[n] = mean over heads
// Everything lives in L2 (<80 MB total) -> bound by L2 atomic throughput.
// ---------------------------------------------------------------------------

typedef float v2f __attribute__((ext_vector_type(2)));
typedef float v8f __attribute__((ext_vector_type(8)));

#define IN_CH   128
#define HIDC    64
#define NHEAD   2

// ---- monotone float<->uint encoding for atomicMax-based segment max --------
__device__ __forceinline__ unsigned fenc(float f) {
    unsigned u = __float_as_uint(f);
    return (u & 0x80000000u) ? ~u : (u | 0x80000000u);
}
__device__ __forceinline__ float fdec(unsigned k) {
    unsigned u = (k & 0x80000000u) ? (k & 0x7FFFFFFFu) : ~k;
    return __uint_as_float(u);
}

// ---- zero-fill (replaces hipMemsetAsync; safe under graph capture) ---------
__global__ void K_zero(float* __restrict__ p, size_t n) {
    size_t i = (size_t)blockIdx.x * blockDim.x + threadIdx.x;
    size_t stride = (size_t)gridDim.x * blockDim.x;
    for (; i < n; i += stride) p[i] = 0.0f;
}

// ---- layer-0 GEMM: h0[N,128] = x[N,128] @ W0[128,128] via f32 WMMA ---------
// One wave -> 16 rows x 128 cols. K loop: 32 steps of V_WMMA_F32_16X16X4_F32.
// A 16x4 layout (ISA 7.12.2): lanes 0-15 hold M=lane, VGPR0/1 = K0/K1;
//                             lanes 16-31 hold M=lane-16, VGPR0/1 = K2/K3.
// B 4x16 mirrored: N across lanes, same K split.
// C/D: lane<16 -> N=lane,   VGPR j = row M=j;
//      lane>=16 -> N=lane-16, VGPR j = row M=j+8.
__global__ void K_gemm0(const float* __restrict__ x, const float* __restrict__ W,
                        float* __restrict__ h0, int nstrips) {
    int wave  = (int)((blockIdx.x * (unsigned)blockDim.x + threadIdx.x) >> 5);
    int lane  = threadIdx.x & 31;
    if (wave >= nstrips) return;
    int row0 = wave * 16;
    int ml   = lane & 15;        // M (for A) / N (for B) within tile
    int khi  = (lane >> 4) * 2;  // K offset selected by lane half

    v8f acc[8];
    v8f z = {0.f, 0.f, 0.f, 0.f, 0.f, 0.f, 0.f, 0.f};
#pragma unroll
    for (int t = 0; t < 8; ++t) acc[t] = z;

    const float* arow = x + (size_t)(row0 + ml) * IN_CH;
    for (int k = 0; k < IN_CH; k += 4) {
        int kb = k + khi;
        v2f a;
        a.x = arow[kb];
        a.y = arow[kb + 1];
        const float* wr0 = W + (size_t)kb * 128;
#pragma unroll
        for (int t = 0; t < 8; ++t) {
            int n = t * 16 + ml;
            v2f b;
            b.x = wr0[n];
            b.y = wr0[128 + n];
            acc[t] = __builtin_amdgcn_wmma_f32_16x16x4_f32(
                false, a, false, b, (short)0, acc[t], false, false);
        }
    }
    int mrow = row0 + (lane >> 4) * 8;
#pragma unroll
    for (int t = 0; t < 8; ++t) {
#pragma unroll
        for (int j = 0; j < 8; ++j)
            h0[(size_t)(mrow + j) * 128 + t * 16 + ml] = acc[t][j];
    }
}

// ---- per-node attention scalars for layer 0 --------------------------------
// a_dst[n,h] = <h0[n,h,:], attn0[h,0:64]> ; a_src[n,h] = <h0[n,h,:], attn0[h,64:128]>
__global__ void K_nalpha0(const float* __restrict__ h0, const float* __restrict__ attn,
                          float* __restrict__ adst, float* __restrict__ asrc, int N) {
    int i = blockIdx.x * blockDim.x + threadIdx.x;
    if (i >= N * NHEAD) return;
    int n = i >> 1, h = i & 1;
    const float* hp = h0 + (size_t)n * 128 + h * 64;
    const float* aL = attn + h * 128;
    const float* aR = aL + 64;
    float s1 = 0.f, s2 = 0.f;
#pragma unroll 8
    for (int c = 0; c < 64; ++c) { float v = hp[c]; s1 += v * aL[c]; s2 += v * aR[c]; }
    adst[i] = s1; asrc[i] = s2;
}

// ---- edge pass 1: leaky-relu + segment max ---------------------------------
__global__ void K_emax(const int* __restrict__ ei,
                       const float* __restrict__ adst, const float* __restrict__ asrc,
                       float* __restrict__ raw, unsigned* __restrict__ maxkey, int E) {
    int i = blockIdx.x * blockDim.x + threadIdx.x;
    if (i >= E * NHEAD) return;
    int e = i >> 1, h = i & 1;
    int r = ei[e], c = ei[E + e];
    float a = adst[r * 2 + h] + asrc[c * 2 + h];
    a = (a > 0.f) ? a : 0.2f * a;
    raw[i] = a;
    atomicMax(&maxkey[r * 2 + h], fenc(a));
}

// ---- edge pass 2: exp + segment sum (overwrites raw with ex) ---------------
__global__ void K_eexp(const int* __restrict__ ei, float* __restrict__ raw_ex,
                       const unsigned* __restrict__ maxkey, float* __restrict__ sum, int E) {
    int i = blockIdx.x * blockDim.x + threadIdx.x;
    if (i >= E * NHEAD) return;
    int e = i >> 1, h = i & 1;
    int r = ei[e];
    float ex = expf(raw_ex[i] - fdec(maxkey[r * 2 + h]));
    raw_ex[i] = ex;
    atomicAdd(&sum[r * 2 + h], ex);
}

// ---- layer-0 aggregation: out0[row,:] += w * h0[col,:] ---------------------
// thread = (edge, 4-channel chunk); 32 chunks cover both heads' 128 channels.
__global__ void K_agg0(const int* __restrict__ ei, const float* __restrict__ ex,
                       const float* __restrict__ sum, const float* __restrict__ h0,
                       float* __restrict__ out0, int E) {
    long long i = (long long)blockIdx.x * blockDim.x + threadIdx.x;
    if (i >= (long long)E * 32) return;
    int e  = (int)(i >> 5);
    int c4 = (int)(i & 31);
    int h  = c4 >> 4;
    int r = ei[e], c = ei[E + e];
    float w = ex[e * 2 + h] / sum[r * 2 + h];
    const float4 v = *(const float4*)(h0 + (size_t)c * 128 + c4 * 4);
    float* op = out0 + (size_t)r * 128 + c4 * 4;
    atomicAdd(op + 0, w * v.x);
    atomicAdd(op + 1, w * v.y);
    atomicAdd(op + 2, w * v.z);
    atomicAdd(op + 3, w * v.w);
}

// ---- head-mean + relu ------------------------------------------------------
__global__ void K_mid(const float* __restrict__ out0, float* __restrict__ hmid, int N) {
    int i = blockIdx.x * blockDim.x + threadIdx.x;
    if (i >= N * HIDC) return;
    int n = i >> 6, c = i & 63;
    float v = 0.5f * (out0[(size_t)n * 128 + c] + out0[(size_t)n * 128 + 64 + c]);
    hmid[i] = (v > 0.f) ? v : 0.f;
}

// ---- layer-1 linear (64x2) + per-node attention scalars --------------------
__global__ void K_node1(const float* __restrict__ hmid, const float* __restrict__ W1,
                        const float* __restrict__ attn1, float* __restrict__ h1,
                        float* __restrict__ adst, float* __restrict__ asrc, int N) {
    int i = blockIdx.x * blockDim.x + threadIdx.x;
    if (i >= N * NHEAD) return;
    int n = i >> 1, h = i & 1;
    const float* hp = hmid + (size_t)n * 64;
    float s = 0.f;
#pragma unroll 8
    for (int c = 0; c < 64; ++c) s += hp[c] * W1[c * 2 + h];
    h1[i]   = s;
    adst[i] = s * attn1[h * 2 + 0];
    asrc[i] = s * attn1[h * 2 + 1];
}

// ---- layer-1 aggregation ---------------------------------------------------
__global__ void K_agg1(const int* __restrict__ ei, const float* __restrict__ ex,
                       const float* __restrict__ sum, const float* __restrict__ h1,
                       float* __restrict__ out1, int E) {
    int i = blockIdx.x * blockDim.x + threadIdx.x;
    if (i >= E * NHEAD) return;
    int e = i >> 1, h = i & 1;
    int r = ei[e], c = ei[E + e];
    float w = ex[i] / sum[r * 2 + h];
    atomicAdd(&out1[r * 2 + h], w * h1[c * 2 + h]);
}

// ---- final: mean over heads -> [N,1] ---------------------------------------
__global__ void K_final(const float* __restrict__ out1, float* __restrict__ out, int N) {
    int n = blockIdx.x * blockDim.x + threadIdx.x;
    if (n >= N) return;
    out[n] = 0.5f * (out1[n * 2] + out1[n * 2 + 1]);
}

// ---------------------------------------------------------------------------
extern "C" void kernel_launch(void* const* d_in, const int* in_sizes, int n_in,
                              void* d_out, int out_size, void* d_ws, size_t ws_size,
                              hipStream_t stream) {
    const float* x     = (const float*)d_in[0];
    const int*   ei    = (const int*)  d_in[1];
    const float* W0    = (const float*)d_in[2];
    const float* attn0 = (const float*)d_in[3];
    const float* W1    = (const float*)d_in[4];
    const float* attn1 = (const float*)d_in[5];
    const int N = in_sizes[0] / IN_CH;
    const int E = in_sizes[1] / 2;

    float* ws = (float*)d_ws;
    size_t o = 0;
    float* h0    = ws + o; o += (size_t)N * 128;   // layer-0 linear output
    float* ex0   = ws + o; o += (size_t)E * 2;     // raw alpha -> ex (layer 0)
    float* adst0 = ws + o; o += (size_t)N * 2;
    float* asrc0 = ws + o; o += (size_t)N * 2;
    float* hmid  = ws + o; o += (size_t)N * 64;
    float* h1    = ws + o; o += (size_t)N * 2;
    float* adst1 = ws + o; o += (size_t)N * 2;
    float* asrc1 = ws + o; o += (size_t)N * 2;
    float* ex1   = ws + o; o += (size_t)E * 2;     // raw alpha -> ex (layer 1)
    // --- contiguous zero-initialized block (accumulators + max keys) ---
    float*    zbase   = ws + o;
    unsigned* maxkey0 = (unsigned*)(ws + o); o += (size_t)N * 2;
    float*    sum0    = ws + o;              o += (size_t)N * 2;
    unsigned* maxkey1 = (unsigned*)(ws + o); o += (size_t)N * 2;
    float*    sum1    = ws + o;              o += (size_t)N * 2;
    float*    out1    = ws + o;              o += (size_t)N * 2;
    float*    out0    = ws + o;              o += (size_t)N * 128;
    size_t zcount = (size_t)(ws + o - zbase);

    // zero accumulators (maxkey==0 decodes below every real float -> acts as -inf)
    K_zero<<<4096, 256, 0, stream>>>(zbase, zcount);

    // layer 0
    int nstrips = (N + 15) / 16;                       // 50000/16 = 3125 exact
    int gblocks = (nstrips * 32 + 127) / 128;          // 4 waves / block
    K_gemm0<<<gblocks, 128, 0, stream>>>(x, W0, h0, nstrips);
    K_nalpha0<<<(N * 2 + 255) / 256, 256, 0, stream>>>(h0, attn0, adst0, asrc0, N);
    K_emax<<<(E * 2 + 255) / 256, 256, 0, stream>>>(ei, adst0, asrc0, ex0, maxkey0, E);
    K_eexp<<<(E * 2 + 255) / 256, 256, 0, stream>>>(ei, ex0, maxkey0, sum0, E);
    long long aggN = (long long)E * 32;
    K_agg0<<<(unsigned)((aggN + 255) / 256), 256, 0, stream>>>(ei, ex0, sum0, h0, out0, E);
    K_mid<<<(N * 64 + 255) / 256, 256, 0, stream>>>(out0, hmid, N);

    // layer 1
    K_node1<<<(N * 2 + 255) / 256, 256, 0, stream>>>(hmid, W1, attn1, h1, adst1, asrc1, N);
    K_emax<<<(E * 2 + 255) / 256, 256, 0, stream>>>(ei, adst1, asrc1, ex1, maxkey1, E);
    K_eexp<<<(E * 2 + 255) / 256, 256, 0, stream>>>(ei, ex1, maxkey1, sum1, E);
    K_agg1<<<(E * 2 + 255) / 256, 256, 0, stream>>>(ei, ex1, sum1, h1, out1, E);
    K_final<<<(N + 255) / 256, 256, 0, stream>>>(out1, (float*)d_out, N);
}